// Encoder_conv_17798344474724
// MI455X (gfx1250) — compile-verified
//
#include <hip/hip_runtime.h>
#include <hip/hip_bf16.h>
#include <math.h>

// ---------------------------------------------------------------------------
// Encoder_conv forward for MI455X (gfx1250, wave32, WMMA).
//
// Sizes: NLAYERS=3, H=NHID=256, NINP=128, K=2, B=32, W=512.
// All GEMMs use V_WMMA_F32_16X16X4_F32 (fp32 MAC array; inputs are fp32 and
// the 512-step recurrence punishes low precision).
//
// Pipeline (all on `stream`, stream-ordered):
//   1) emb_kernel  : x0 = emb_w @ input + emb_b          (WMMA GEMM)
//   2) per layer l : ig_kernel  -> ig[b,4H,W] = sum_j conv_j(x_list[j]) (WMMA)
//                    scan_kernel-> 512-step LSTM scan, one 32-wave workgroup,
//                                  hx in LDS, cx/gates in WMMA register frags,
//                                  attention dot via ds_add_f32 LDS atomics
//   3) gather_kernel: pack (x[...,-2:], nc[...,-2:], attns) into d_out
// ---------------------------------------------------------------------------

typedef float v2f __attribute__((ext_vector_type(2)));
typedef float v8f __attribute__((ext_vector_type(8)));

#define NLAYERS 3
#define NH      256
#define NI      128
#define BB      32
#define WW      512
#define G4      1024            // 4*NH
#define XSZ     (BB * NH * WW)  // 4,194,304 floats per x_list entry

static __device__ __forceinline__ v8f wmma4(v2f a, v2f b, v8f c) {
  // D = A(16x4 f32) x B(4x16 f32) + C(16x16 f32)
  return __builtin_amdgcn_wmma_f32_16x16x4_f32(
      /*neg_a=*/false, a, /*neg_b=*/false, b,
      /*c_mod=*/(short)0, c, /*reuse_a=*/false, /*reuse_b=*/false);
}

static __device__ __forceinline__ float sigmoidf_(float x) {
  return 1.0f / (1.0f + __expf(-x));
}

// ---------------------------------------------------------------------------
// x0[b, m, w] = sum_c emb_w[m,c] * input[b,c,w] + emb_b[m]
// One 16x16 output tile per wave.  Tiles: B(32) x Mt(16) x Nt(32) = 16384.
// ---------------------------------------------------------------------------
__global__ void __launch_bounds__(256) emb_kernel(
    const float* __restrict__ inp, const float* __restrict__ ew,
    const float* __restrict__ eb, float* __restrict__ x0) {
  const int lane = threadIdx.x & 31;
  const int ln = lane & 15, kh = lane >> 4;
  int tile = blockIdx.x * 8 + (threadIdx.x >> 5);
  const int nt = tile & 31;
  const int mt = (tile >> 5) & 15;
  const int b  = tile >> 9;
  const int m0 = mt * 16, n0 = nt * 16;

  const float* Am = ew + (size_t)(m0 + ln) * NI;            // A[m][k]
  const float* Bm = inp + (size_t)b * NI * WW + n0 + ln;    // B[k][n] = Bm[k*W]
  v8f acc = {};
  for (int k0 = 0; k0 < NI; k0 += 4) {
    const int ka = k0 + 2 * kh;                 // frag k = v + 2*kh
    v2f a, bv;
    a.x  = Am[ka];           a.y  = Am[ka + 1];
    bv.x = Bm[(size_t)ka * WW]; bv.y = Bm[(size_t)(ka + 1) * WW];
    acc = wmma4(a, bv, acc);
  }
  float* Cm = x0 + (size_t)b * NH * WW + n0 + ln;
  for (int r = 0; r < 8; ++r) {
    const int m = m0 + r + 8 * kh;              // C frag row
    Cm[(size_t)m * WW] = acc[r] + eb[m];
  }
}

// ---------------------------------------------------------------------------
// ig[b, o, w] = sum_{j<=layer} ( Wj[:,:,0] @ xshift_j + Wj[:,:,1] @ x_j ) + bias
// xshift[b,c,w] = x_j[b,c,w-1] for w>0, hidden[j][b,c,1] for w==0.
// Tiles: B(32) x Ot(64) x Wt(32) = 65536, one per wave.
// ---------------------------------------------------------------------------
__global__ void __launch_bounds__(256) ig_kernel(
    const float* __restrict__ xbase, const float* __restrict__ hidden,
    const float* __restrict__ cw, const float* __restrict__ cb,
    float* __restrict__ ig, int layer, int off) {
  const int lane = threadIdx.x & 31;
  const int ln = lane & 15, kh = lane >> 4;
  int tile = blockIdx.x * 8 + (threadIdx.x >> 5);
  const int wt = tile & 31;
  const int ot = (tile >> 5) & 63;
  const int b  = tile >> 11;
  const int o0 = ot * 16, w0 = wt * 16;
  const int wcol = w0 + ln;

  v8f acc = {};
  for (int j = 0; j <= layer; ++j) {
    const float* x   = xbase + (size_t)j * XSZ + (size_t)b * NH * WW;
    const float* wgt = cw + (size_t)(off + j) * G4 * NH * 2
                          + (size_t)(o0 + ln) * NH * 2;      // [c*2 + tap]
    const float* hid = hidden + (size_t)(j * BB + b) * NH * 2;  // [c*2 + 1]
    for (int c0 = 0; c0 < NH; c0 += 4) {
      const int ka = c0 + 2 * kh;
      // tap 1: current column
      v2f a1, b1;
      a1.x = wgt[ka * 2 + 1];        a1.y = wgt[(ka + 1) * 2 + 1];
      b1.x = x[(size_t)ka * WW + wcol];
      b1.y = x[(size_t)(ka + 1) * WW + wcol];
      acc = wmma4(a1, b1, acc);
      // tap 0: previous column (hidden state at the left boundary)
      v2f a0, b0;
      a0.x = wgt[ka * 2];            a0.y = wgt[(ka + 1) * 2];
      if (wcol == 0) {
        b0.x = hid[ka * 2 + 1];      b0.y = hid[(ka + 1) * 2 + 1];
      } else {
        b0.x = x[(size_t)ka * WW + wcol - 1];
        b0.y = x[(size_t)(ka + 1) * WW + wcol - 1];
      }
      acc = wmma4(a0, b0, acc);       // EXEC reconverged here
    }
  }
  for (int r = 0; r < 8; ++r) {
    const int o = o0 + r + 8 * kh;
    float bias = 0.0f;
    for (int j = 0; j <= layer; ++j) bias += cb[(size_t)(off + j) * G4 + o];
    ig[(size_t)b * G4 * WW + (size_t)o * WW + w0 + ln] = acc[r] + bias;
  }
}

// ---------------------------------------------------------------------------
// 512-step LSTM scan for one layer.  Single 1024-thread (32-wave) workgroup.
// Wave w owns (mt = w&1) b-tile and (ht = w>>1) h-tile, and computes the
// i/f/c/o gate tiles for its cells entirely in WMMA C fragments; cx persists
// in registers.  hx lives in LDS as the shared WMMA A operand.
// ---------------------------------------------------------------------------
__global__ void __launch_bounds__(1024) scan_kernel(
    const float* __restrict__ ig, const float* __restrict__ rw,
    const float* __restrict__ rb, const float* __restrict__ aw,
    const float* __restrict__ hidden, const float* __restrict__ context,
    float* __restrict__ hx_out, float* __restrict__ cx_save,
    float* __restrict__ attn_save, int layer) {
  __shared__ float hx_lds[BB * NH];     // 32 KB: hx[b][h]
  __shared__ float a_lds[BB];           // attention dot partials

  const int tid  = threadIdx.x;
  const int lane = tid & 31, wave = tid >> 5;
  const int ln = lane & 15, kh = lane >> 4;
  const int m0 = (wave & 1) * 16;       // b-tile base
  const int h0 = (wave >> 1) * 16;      // h-tile base
  const int hcol = h0 + ln;

  // init hx (hidden[layer+1][:,:,K-1]) into LDS
  for (int idx = tid; idx < BB * NH; idx += 1024) {
    const int b = idx >> 8, h = idx & 255;
    hx_lds[idx] = hidden[(size_t)((layer + 1) * BB + b) * NH * 2 + h * 2 + 1];
  }
  // init cx (context[layer][:,:,K-1]) into register fragment
  v8f cx;
  for (int r = 0; r < 8; ++r) {
    const int b = m0 + r + 8 * kh;
    cx[r] = context[(size_t)(layer * BB + b) * NH * 2 + hcol * 2 + 1];
  }
  const float myaw = aw[hcol];
  // recurrent-weight row pointers for the 4 gate tiles: B[k][n] = rw[n*NH + k]
  const float* rg0 = rw + (size_t)(0 * NH + hcol) * NH;
  const float* rg1 = rw + (size_t)(1 * NH + hcol) * NH;
  const float* rg2 = rw + (size_t)(2 * NH + hcol) * NH;
  const float* rg3 = rw + (size_t)(3 * NH + hcol) * NH;
  __syncthreads();

  for (int t = 0; t < WW; ++t) {
    v8f g0 = {}, g1 = {}, g2 = {}, g3 = {};
    const float* hxA = hx_lds + (m0 + ln) * NH;     // A[m][k]
    for (int k0 = 0; k0 < NH; k0 += 4) {
      const int ka = k0 + 2 * kh;
      v2f a; a.x = hxA[ka]; a.y = hxA[ka + 1];
      v2f b0; b0.x = rg0[ka]; b0.y = rg0[ka + 1];  g0 = wmma4(a, b0, g0);
      v2f b1; b1.x = rg1[ka]; b1.y = rg1[ka + 1];  g1 = wmma4(a, b1, g1);
      v2f b2; b2.x = rg2[ka]; b2.y = rg2[ka + 1];  g2 = wmma4(a, b2, g2);
      v2f b3; b3.x = rg3[ka]; b3.y = rg3[ka + 1];  g3 = wmma4(a, b3, g3);
    }
    // gates = WMMA + ig[:, :, t] + rb ; LSTM nonlinearity in registers
    v8f hx;
    for (int r = 0; r < 8; ++r) {
      const int b = m0 + r + 8 * kh;
      const float* igb = ig + (size_t)b * G4 * WW + t;
      const float iG = g0[r] + igb[(size_t)(0 * NH + hcol) * WW] + rb[0 * NH + hcol];
      const float fG = g1[r] + igb[(size_t)(1 * NH + hcol) * WW] + rb[1 * NH + hcol];
      const float cG = g2[r] + igb[(size_t)(2 * NH + hcol) * WW] + rb[2 * NH + hcol];
      const float oG = g3[r] + igb[(size_t)(3 * NH + hcol) * WW] + rb[3 * NH + hcol];
      const float cn = sigmoidf_(fG) * cx[r] + sigmoidf_(iG) * tanhf(cG);
      cx[r] = cn;
      hx[r] = sigmoidf_(oG) * tanhf(cn);
    }
    // attention dot: a[b] = sum_h hx[b,h]*aw[h]  (ds_add_f32 LDS atomics)
    __syncthreads();
    if (tid < BB) a_lds[tid] = 0.0f;
    __syncthreads();
    for (int r = 0; r < 8; ++r) {
      const int b = m0 + r + 8 * kh;
      atomicAdd(&a_lds[b], hx[r] * myaw);
    }
    __syncthreads();
    // scale (except last step), publish hx to LDS + global, save cx/attn
    for (int r = 0; r < 8; ++r) {
      const int b = m0 + r + 8 * kh;
      const float av = sigmoidf_(a_lds[b]);
      const float hv = (t < WW - 1) ? hx[r] * av : hx[r];
      hx_lds[b * NH + hcol] = hv;
      hx_out[(size_t)b * NH * WW + (size_t)hcol * WW + t] = hv;
      if (t >= WW - 2) cx_save[((size_t)b * NH + hcol) * 2 + (t - (WW - 2))] = cx[r];
    }
    if (tid < BB && t < WW - 1)
      attn_save[(size_t)tid * (WW - 1) + t] = sigmoidf_(a_lds[tid]);
    __syncthreads();
  }
}

// ---------------------------------------------------------------------------
// Pack outputs: x[:, :, :, -2:] (4,32,256,2) | nc (3,32,256,2) | attns (3,32,511)
// ---------------------------------------------------------------------------
__global__ void gather_kernel(const float* __restrict__ x,
                              const float* __restrict__ cxs,
                              const float* __restrict__ ats,
                              float* __restrict__ out, int total) {
  const int i = blockIdx.x * blockDim.x + threadIdx.x;
  if (i >= total) return;
  const int S1 = 4 * BB * NH * 2;   // 65536
  const int S2 = NLAYERS * BB * NH * 2;  // 49152
  if (i < S1) {
    const int k = i & 1, h = (i >> 1) & 255, b = (i >> 9) & 31, s = i >> 14;
    out[i] = x[(size_t)((s * BB + b) * NH + h) * WW + (WW - 2) + k];
  } else if (i < S1 + S2) {
    out[i] = cxs[i - S1];
  } else {
    out[i] = ats[i - S1 - S2];
  }
}

// ---------------------------------------------------------------------------
extern "C" void kernel_launch(void* const* d_in, const int* in_sizes, int n_in,
                              void* d_out, int out_size, void* d_ws, size_t ws_size,
                              hipStream_t stream) {
  const float* input   = (const float*)d_in[0];
  const float* hidden  = (const float*)d_in[1];
  const float* context = (const float*)d_in[2];
  const float* emb_w   = (const float*)d_in[3];
  const float* emb_b   = (const float*)d_in[4];
  const float* conv_w  = (const float*)d_in[5];
  const float* conv_b  = (const float*)d_in[6];
  const float* rec_w   = (const float*)d_in[7];
  const float* rec_b   = (const float*)d_in[8];
  const float* attn_w  = (const float*)d_in[9];
  float* out = (float*)d_out;

  // workspace layout (floats): x_list[4] | ig | cx_save | attn_save (~135 MB)
  float* x   = (float*)d_ws;
  float* ig  = x + (size_t)4 * XSZ;
  float* cxs = ig + (size_t)BB * G4 * WW;
  float* ats = cxs + (size_t)NLAYERS * BB * NH * 2;

  // x0 = embedding
  emb_kernel<<<2048, 256, 0, stream>>>(input, emb_w, emb_b, x);

  int off = 0;
  for (int l = 0; l < NLAYERS; ++l) {
    ig_kernel<<<8192, 256, 0, stream>>>(x, hidden, conv_w, conv_b, ig, l, off);
    scan_kernel<<<1, 1024, 0, stream>>>(
        ig, rec_w + (size_t)l * G4 * NH, rec_b + (size_t)l * G4,
        attn_w + (size_t)l * NH, hidden, context,
        x + (size_t)(l + 1) * XSZ,
        cxs + (size_t)l * BB * NH * 2,
        ats + (size_t)l * BB * (WW - 1), l);
    off += l + 1;
  }

  const int total = 4 * BB * NH * 2 + NLAYERS * BB * NH * 2 + NLAYERS * BB * (WW - 1);
  gather_kernel<<<(total + 255) / 256, 256, 0, stream>>>(x, cxs, ats, out, total);
}